// Net_90074054132246
// MI455X (gfx1250) — compile-verified
//
#include <hip/hip_runtime.h>
#include <hip/hip_bf16.h>

typedef __attribute__((ext_vector_type(2))) float v2f;
typedef __attribute__((ext_vector_type(8))) float v8f;

#define N_NODES 100000
#define N_EDGES 1600000
#define NB16    (N_NODES / 16)   // 6250 exact 16-row tiles, no remainder

__device__ __forceinline__ v2f ld2(const float* p) {
  return *reinterpret_cast<const v2f*>(p);
}

// ---------------------------------------------------------------- utilities

__global__ void zero_f32(float* __restrict__ p, int n) {
  int i = blockIdx.x * blockDim.x + threadIdx.x;
  if (i < n) p[i] = 0.0f;
}

__global__ void degree_kernel(const int* __restrict__ dst, float* __restrict__ deg, int E) {
  int e = blockIdx.x * blockDim.x + threadIdx.x;
  if (e < E) atomicAdd(&deg[dst[e]], 1.0f);
}

__global__ void dinv_kernel(const float* __restrict__ deg, float* __restrict__ dinv, int n) {
  int i = blockIdx.x * blockDim.x + threadIdx.x;
  if (i < n) dinv[i] = 1.0f / sqrtf(deg[i] + 1.0f);
}

// x [N,7] -> x8 [N,8] zero-padded
__global__ void pad_x_kernel(const float* __restrict__ x, float* __restrict__ x8, int n) {
  int idx = blockIdx.x * blockDim.x + threadIdx.x;
  if (idx >= n * 8) return;
  int i = idx >> 3, t = idx & 7;
  x8[idx] = (t < 7) ? x[i * 7 + t] : 0.0f;
}

// W [rows,7] -> W8 [rows,8] zero-padded
__global__ void pad_w_kernel(const float* __restrict__ W, float* __restrict__ W8, int rows) {
  int idx = blockIdx.x * blockDim.x + threadIdx.x;
  if (idx >= rows * 8) return;
  int r = idx >> 3, t = idx & 7;
  W8[idx] = (t < 7) ? W[r * 7 + t] : 0.0f;
}

// One thread per (edge, feature): agg[dst] += h[src]
__global__ void scatter_sum(const float* __restrict__ h,
                            const int* __restrict__ src,
                            const int* __restrict__ dst,
                            float* __restrict__ agg, int E, int D) {
  int idx = blockIdx.x * blockDim.x + threadIdx.x;
  if (idx >= E * D) return;
  int e = idx / D;
  int t = idx - e * D;
  int s = src[e];
  int d = dst[e];
  atomicAdd(&agg[d * D + t], h[s * D + t]);
}

// -------------------------------------------------- SAGE dense layer (WMMA)
// h_out = relu((agg/denom) @ Wl^T + bl + h_in @ Wr^T)
// One wave computes a full 16 x DOUT strip (NT = DOUT/16 accumulator tiles).
// f32 WMMA 16x16x4: A lanes 0-15 hold K=k0,k0+1 ; lanes 16-31 hold k0+2,k0+3.
// C/D: VGPR r, lanes 0-15 -> M=r ; lanes 16-31 -> M=8+r ; N = lane&15.
// N_NODES % 16 == 0 and DIN % 4 == 0, so there are no guards anywhere.
template<int DIN, int DOUT>
__global__ void sage_dense_wmma(const float* __restrict__ hin,
                                const float* __restrict__ agg,
                                const float* __restrict__ deg,
                                const float* __restrict__ Wl,
                                const float* __restrict__ bl,
                                const float* __restrict__ Wr,
                                float* __restrict__ hout) {
  constexpr int NT = DOUT / 16;
  const int lane  = threadIdx.x;        // 32 threads per block = 1 wave
  const int r16   = lane & 15;
  const int khalf = lane >> 4;
  const int m0    = blockIdx.x * 16;
  const int row   = m0 + r16;

  const float dg  = deg[row];
  const float inv = 1.0f / (dg > 1.0f ? dg : 1.0f);

  v8f acc[NT];
#pragma unroll
  for (int nt = 0; nt < NT; nt++) acc[nt] = (v8f){};

  const float* arow = agg + (size_t)row * DIN;
  const float* hrow = hin + (size_t)row * DIN;

  // term 1: (agg * inv) @ Wl^T
#pragma unroll 2
  for (int k0 = 0; k0 < DIN; k0 += 4) {
    const int kb = k0 + khalf * 2;
    v2f a = ld2(arow + kb);
    a.x *= inv; a.y *= inv;
#pragma unroll
    for (int nt = 0; nt < NT; nt++) {
      v2f b = ld2(Wl + (nt * 16 + r16) * DIN + kb);   // B[k][n] = Wl[n][k]
      acc[nt] = __builtin_amdgcn_wmma_f32_16x16x4_f32(false, a, false, b,
                                                      (short)0, acc[nt], false, false);
    }
  }
  // term 2: h_in @ Wr^T
#pragma unroll 2
  for (int k0 = 0; k0 < DIN; k0 += 4) {
    const int kb = k0 + khalf * 2;
    v2f a = ld2(hrow + kb);
#pragma unroll
    for (int nt = 0; nt < NT; nt++) {
      v2f b = ld2(Wr + (nt * 16 + r16) * DIN + kb);
      acc[nt] = __builtin_amdgcn_wmma_f32_16x16x4_f32(false, a, false, b,
                                                      (short)0, acc[nt], false, false);
    }
  }

  const int mrow = m0 + khalf * 8;
#pragma unroll
  for (int nt = 0; nt < NT; nt++) {
    const float bias = bl[nt * 16 + r16];
#pragma unroll
    for (int r = 0; r < 8; r++) {
      float v = acc[nt][r] + bias;
      hout[(size_t)(mrow + r) * DOUT + nt * 16 + r16] = v > 0.0f ? v : 0.0f;
    }
  }
}

// -------------------------------------------------- plain GEMM (WMMA): out = h @ W^T
template<int DIN, int DOUT>
__global__ void gemm_wmma(const float* __restrict__ hin,
                          const float* __restrict__ W,
                          float* __restrict__ out) {
  constexpr int NT = DOUT / 16;
  const int lane  = threadIdx.x;
  const int r16   = lane & 15;
  const int khalf = lane >> 4;
  const int m0    = blockIdx.x * 16;
  const int row   = m0 + r16;

  v8f acc[NT];
#pragma unroll
  for (int nt = 0; nt < NT; nt++) acc[nt] = (v8f){};

  const float* hrow = hin + (size_t)row * DIN;

#pragma unroll 2
  for (int k0 = 0; k0 < DIN; k0 += 4) {
    const int kb = k0 + khalf * 2;
    v2f a = ld2(hrow + kb);
#pragma unroll
    for (int nt = 0; nt < NT; nt++) {
      v2f b = ld2(W + (nt * 16 + r16) * DIN + kb);
      acc[nt] = __builtin_amdgcn_wmma_f32_16x16x4_f32(false, a, false, b,
                                                      (short)0, acc[nt], false, false);
    }
  }

  const int mrow = m0 + khalf * 8;
#pragma unroll
  for (int nt = 0; nt < NT; nt++) {
#pragma unroll
    for (int r = 0; r < 8; r++) {
      out[(size_t)(mrow + r) * DOUT + nt * 16 + r16] = acc[nt][r];
    }
  }
}

// -------------------------------------------------- GCN tail
// out[i] = hg[i] * dinv_i^2 + bg   (self-loop term + bias)
__global__ void gcn_init(const float* __restrict__ hg,
                         const float* __restrict__ dinv,
                         const float* __restrict__ bg,
                         float* __restrict__ out, int n) {
  int idx = blockIdx.x * blockDim.x + threadIdx.x;
  if (idx >= n * 16) return;
  int i = idx >> 4;
  int t = idx & 15;
  float di = dinv[i];
  out[idx] = hg[idx] * di * di + bg[t];
}

// out[dst] += hg[src] * dinv[src]*dinv[dst]
__global__ void gcn_scatter(const float* __restrict__ hg,
                            const float* __restrict__ dinv,
                            const int* __restrict__ src,
                            const int* __restrict__ dst,
                            float* __restrict__ out, int E) {
  int idx = blockIdx.x * blockDim.x + threadIdx.x;
  if (idx >= E * 16) return;
  int e = idx >> 4;
  int t = idx & 15;
  int s = src[e];
  int d = dst[e];
  float w = dinv[s] * dinv[d];
  atomicAdd(&out[d * 16 + t], hg[s * 16 + t] * w);
}

// ---------------------------------------------------------------- launcher

extern "C" void kernel_launch(void* const* d_in, const int* in_sizes, int n_in,
                              void* d_out, int out_size, void* d_ws, size_t ws_size,
                              hipStream_t stream) {
  (void)in_sizes; (void)n_in; (void)out_size; (void)ws_size;

  const float* x   = (const float*)d_in[0];
  const int*   ei  = (const int*)d_in[1];
  const int*   src = ei;             // edge_index[0]
  const int*   dst = ei + N_EDGES;   // edge_index[1]

  const float* Wl[6]; const float* bl[6]; const float* Wr[6];
  for (int i = 0; i < 6; i++) {
    Wl[i] = (const float*)d_in[2 + 3 * i];
    bl[i] = (const float*)d_in[3 + 3 * i];
    Wr[i] = (const float*)d_in[4 + 3 * i];
  }
  const float* Wg = (const float*)d_in[20];
  const float* bg = (const float*)d_in[21];
  float* out = (float*)d_out;

  // workspace (floats):
  // deg[N] | dinv[N] | agg[N*128] | hA[N*128] | hB[N*128] | x8[N*8] | Wl8[256] | Wr8[256]
  float* ws   = (float*)d_ws;
  float* deg  = ws;
  float* dinv = deg  + N_NODES;
  float* agg  = dinv + N_NODES;
  float* hA   = agg  + (size_t)N_NODES * 128;
  float* hB   = hA   + (size_t)N_NODES * 128;
  float* x8   = hB   + (size_t)N_NODES * 128;
  float* Wl8  = x8   + (size_t)N_NODES * 8;
  float* Wr8  = Wl8  + 32 * 8;

  // in-degree (SAGE mean denom) and GCN norm 1/sqrt(deg+1)
  zero_f32<<<(N_NODES + 255) / 256, 256, 0, stream>>>(deg, N_NODES);
  degree_kernel<<<(N_EDGES + 255) / 256, 256, 0, stream>>>(dst, deg, N_EDGES);
  dinv_kernel<<<(N_NODES + 255) / 256, 256, 0, stream>>>(deg, dinv, N_NODES);

  // pad din=7 -> 8 so every layer has compile-time DIN % 4 == 0
  pad_x_kernel<<<(N_NODES * 8 + 255) / 256, 256, 0, stream>>>(x, x8, N_NODES);
  pad_w_kernel<<<(32 * 8 + 255) / 256, 256, 0, stream>>>(Wl[0], Wl8, 32);
  pad_w_kernel<<<(32 * 8 + 255) / 256, 256, 0, stream>>>(Wr[0], Wr8, 32);

#define SAGE_LAYER(DIN, DOUT, HIN, WLP, BLP, WRP, HOUT)                                   \
  do {                                                                                    \
    zero_f32<<<(N_NODES * (DIN) + 255) / 256, 256, 0, stream>>>(agg, N_NODES * (DIN));    \
    scatter_sum<<<(N_EDGES * (DIN) + 255) / 256, 256, 0, stream>>>(                       \
        (HIN), src, dst, agg, N_EDGES, (DIN));                                            \
    sage_dense_wmma<DIN, DOUT><<<NB16, 32, 0, stream>>>(                                  \
        (HIN), agg, deg, (WLP), (BLP), (WRP), (HOUT));                                    \
  } while (0)

  SAGE_LAYER(8,   32,  x8, Wl8,   bl[0], Wr8,   hA);
  SAGE_LAYER(32,  64,  hA, Wl[1], bl[1], Wr[1], hB);
  SAGE_LAYER(64,  128, hB, Wl[2], bl[2], Wr[2], hA);
  SAGE_LAYER(128, 64,  hA, Wl[3], bl[3], Wr[3], hB);
  SAGE_LAYER(64,  32,  hB, Wl[4], bl[4], Wr[4], hA);
  SAGE_LAYER(32,  32,  hA, Wl[5], bl[5], Wr[5], hB);
#undef SAGE_LAYER

  // GCN: hg = h @ Wg^T  (32 -> 16); reuse agg buffer for hg.
  float* hg = agg;
  gemm_wmma<32, 16><<<NB16, 32, 0, stream>>>(hB, Wg, hg);

  gcn_init<<<(N_NODES * 16 + 255) / 256, 256, 0, stream>>>(hg, dinv, bg, out, N_NODES);
  gcn_scatter<<<(N_EDGES * 16 + 255) / 256, 256, 0, stream>>>(hg, dinv, src, dst, out, N_EDGES);
}